// HierarchicalCubeMap_54030688584364
// MI455X (gfx1250) — compile-verified
//
#include <hip/hip_runtime.h>
#include <hip/hip_bf16.h>
#include <math.h>

// ---------------------------------------------------------------------------
// HierarchicalCubeMap on MI455X (gfx1250, wave32)
//  Phase 1: build mip pyramids for both cubemaps into d_ws (16 small kernels,
//           division-free indexing, vectorizable contiguous loads/stores)
//  Phase 2: per-ray trilinear cubemap sampling; hottest mip levels staged
//           into LDS with the Tensor Data Mover (tensor_load_to_lds +
//           s_wait_tensorcnt) and addressed via a generic-pointer level table.
// ---------------------------------------------------------------------------

typedef unsigned int u32x4 __attribute__((ext_vector_type(4)));
typedef int          i32x8 __attribute__((ext_vector_type(8)));
typedef int          i32x4 __attribute__((ext_vector_type(4)));

#define B_TOTAL   (1u << 20)

// ---- d_ws layout (float offsets) ------------------------------------------
// tex0 (128^2 base): levels 1..7 contiguous
#define WS0_L1   0          // 64^2 : 73728
#define WS0_L2   73728      // 32^2 : 18432
#define WS0_L3   92160      // 16^2 .. 1^2 : 6138 (staged to LDS)
// tex1 (512^2 base): levels 1..9 contiguous
#define WS1_BASE 98298
#define WS1_L2   (WS1_BASE + 1179648)
#define WS1_L3   (WS1_BASE + 1474560)
#define WS1_L4   (WS1_BASE + 1548288)
#define WS1_L5   (WS1_BASE + 1566720) // 16^2 .. 1^2 : 6138 (staged to LDS)
#define STAGE_N  6138

// ---------------------------------------------------------------------------
// 2x2 box-filter downsample, one thread per OUTPUT TEXEL (3 channels each).
// Rd = 1 << rdBits is the destination resolution; N = 6*Rd*Rd texels.
// Division-free index decode; contiguous 24B source spans per row.
// ---------------------------------------------------------------------------
__global__ void downsample_kernel(const float* __restrict__ src,
                                  float* __restrict__ dst,
                                  int rdBits, int N) {
  int idx = blockIdx.x * blockDim.x + threadIdx.x;
  if (idx >= N) return;
  int Rd  = 1 << rdBits;
  int x   = idx & (Rd - 1);
  int y   = (idx >> rdBits) & (Rd - 1);
  int f   = idx >> (2 * rdBits);
  int Rs  = Rd << 1;
  int rs3 = Rs * 3;
  const float* p0 = src + ((f * Rs + 2 * y) * Rs + 2 * x) * 3; // row 2y  : 6 floats
  const float* p1 = p0 + rs3;                                  // row 2y+1: 6 floats
  float* q = dst + idx * 3;                                    // 3 contiguous floats
#pragma unroll
  for (int c = 0; c < 3; ++c)
    q[c] = 0.25f * (p0[c] + p0[c + 3] + p1[c] + p1[c + 3]);
}

// ---------------------------------------------------------------------------
// TDM: 1-D tile load Global -> LDS (n 4-byte elements). D# per ISA ch.8.
// ---------------------------------------------------------------------------
__device__ __forceinline__ void tdm_load_1d(unsigned lds_addr,
                                            unsigned long long gaddr,
                                            unsigned n) {
  u32x4 g0;
  g0[0] = 1u;                                           // count=1, user mode
  g0[1] = lds_addr;                                     // lds_addr [63:32]
  g0[2] = (unsigned)(gaddr & 0xFFFFFFFFull);            // global_addr[31:0]
  g0[3] = ((unsigned)(gaddr >> 32) & 0x01FFFFFFu)       // global_addr[56:32]
          | 0x80000000u;                                // type=2 ("image")
  i32x8 g1;
  g1[0] = 0x00020000;                                   // wg_mask=0, data_size=4B
  g1[1] = (int)((n & 0xFFFFu) << 16);                   // tensor_dim0[15:0]
  g1[2] = (int)(((n >> 16) & 0xFFFFu) | (1u << 16));    // dim0[31:16] | tensor_dim1=1
  g1[3] = (int)((n & 0xFFFFu) << 16);                   // tile_dim0
  g1[4] = 1;                                            // tile_dim1=1
  g1[5] = (int)n;                                       // tensor_dim0_stride[31:0]
  g1[6] = 0;
  g1[7] = 0;
  i32x4 gz = {0, 0, 0, 0};
#if __clang_major__ >= 23
  i32x8 gz8 = {0, 0, 0, 0, 0, 0, 0, 0};
  __builtin_amdgcn_tensor_load_to_lds(g0, g1, gz, gz, gz8, 0);
#else
  __builtin_amdgcn_tensor_load_to_lds(g0, g1, gz, gz, 0);
#endif
}

// ---------------------------------------------------------------------------
// Bilinear gather (clamp-to-edge), weight folded in, accumulate 3 channels.
// p is a generic pointer (LDS-staged or global level data; flat loads).
// ---------------------------------------------------------------------------
__device__ __forceinline__ void bilin_acc(const float* p, int R, int face,
                                          float u, float v, float w,
                                          float& r, float& g, float& b) {
  float xx = u * (float)R - 0.5f;
  float yy = v * (float)R - 0.5f;
  float xf = floorf(xx), yf = floorf(yy);
  float fx = xx - xf, fy = yy - yf;
  int xi = (int)xf, yi = (int)yf;
  int Rm1 = R - 1;
  int x0 = min(max(xi, 0), Rm1);
  int x1 = min(max(xi + 1, 0), Rm1);
  int y0 = min(max(yi, 0), Rm1);
  int y1 = min(max(yi + 1, 0), Rm1);
  int fb = face * R * R;
  const float* p00 = p + 3 * (fb + y0 * R + x0);
  const float* p01 = p + 3 * (fb + y0 * R + x1);
  const float* p10 = p + 3 * (fb + y1 * R + x0);
  const float* p11 = p + 3 * (fb + y1 * R + x1);
  float w00 = (1.0f - fx) * (1.0f - fy) * w;
  float w01 = fx * (1.0f - fy) * w;
  float w10 = (1.0f - fx) * fy * w;
  float w11 = fx * fy * w;
  r += w00 * p00[0] + w01 * p01[0] + w10 * p10[0] + w11 * p11[0];
  g += w00 * p00[1] + w01 * p01[1] + w10 * p10[1] + w11 * p11[1];
  b += w00 * p00[2] + w01 * p01[2] + w10 * p10[2] + w11 * p11[2];
}

// ---------------------------------------------------------------------------
// Per-ray sampling kernel. 1024 threads (32 wave32 waves) per block.
// ---------------------------------------------------------------------------
__global__ __launch_bounds__(1024) void hcubemap_sample(
    const float* __restrict__ vd,  const float* __restrict__ sa,
    const float* __restrict__ nu,  const float* __restrict__ bg0,
    const float* __restrict__ bg1, const float* __restrict__ ws,
    const float* __restrict__ brp, const float* __restrict__ mulp,
    const float* __restrict__ mbp, float* __restrict__ out, unsigned n) {
  __shared__ float s_tex[2 * STAGE_N];           // 49104 B staged mip data
  __shared__ const float* s_lvl0[8];             // level base pointers tex0
  __shared__ const float* s_lvl1[10];            // level base pointers tex1

  if (threadIdx.x < 32) {                        // wave 0 only
    if (threadIdx.x == 0) {
      s_lvl0[0] = bg0;
      s_lvl0[1] = ws + WS0_L1;
      s_lvl0[2] = ws + WS0_L2;
      s_lvl0[3] = s_tex;                         // 16^2
      s_lvl0[4] = s_tex + 4608;                  // 8^2
      s_lvl0[5] = s_tex + 5760;                  // 4^2
      s_lvl0[6] = s_tex + 6048;                  // 2^2
      s_lvl0[7] = s_tex + 6120;                  // 1^2
      s_lvl1[0] = bg1;
      s_lvl1[1] = ws + WS1_BASE;
      s_lvl1[2] = ws + WS1_L2;
      s_lvl1[3] = ws + WS1_L3;
      s_lvl1[4] = ws + WS1_L4;
      s_lvl1[5] = s_tex + STAGE_N;               // 16^2
      s_lvl1[6] = s_tex + STAGE_N + 4608;        // 8^2
      s_lvl1[7] = s_tex + STAGE_N + 5760;        // 4^2
      s_lvl1[8] = s_tex + STAGE_N + 6048;        // 2^2
      s_lvl1[9] = s_tex + STAGE_N + 6120;        // 1^2
    }
    unsigned lds0 = (unsigned)(size_t)(const void*)&s_tex[0];
    unsigned lds1 = (unsigned)(size_t)(const void*)&s_tex[STAGE_N];
    tdm_load_1d(lds0, (unsigned long long)(size_t)(ws + WS0_L3), STAGE_N);
    tdm_load_1d(lds1, (unsigned long long)(size_t)(ws + WS1_L5), STAGE_N);
    __builtin_amdgcn_s_wait_tensorcnt(0);
  }
  __syncthreads();

  unsigned i = blockIdx.x * blockDim.x + threadIdx.x;
  if (i >= n) return;

  const float* vp = vd + 3 * i;
  float x = vp[0], y = vp[1], z = vp[2];
  float ax = fabsf(x), ay = fabsf(y), az = fabsf(z);
  bool is_x = (ax >= ay) && (ax >= az);
  bool is_y = !is_x && (ay >= az);
  int face; float ma, sc, tc;
  if (is_x)      { face = (x >= 0.f) ? 0 : 1; ma = ax; sc = (x >= 0.f) ? -z : z; tc = -y; }
  else if (is_y) { face = (y >= 0.f) ? 2 : 3; ma = ay; sc = x;  tc = (y >= 0.f) ? z : -z; }
  else           { face = (z >= 0.f) ? 4 : 5; ma = az; sc = (z >= 0.f) ? x : -x; tc = -y; }
  float inv_ma = 1.0f / ma;
  float u = 0.5f * (sc * inv_ma + 1.0f);
  float t = 0.5f * (tc * inv_ma + 1.0f);

  // miplevel = (sa - log(saTexel)) / log(4) / 2 + mipbias + 0.5*noise
  float saTexel = inv_ma * (1.0f / (512.0f * 512.0f));
  float mip = (sa[i] - logf(saTexel)) * 0.36067376022224085f + mbp[0] + 0.5f * nu[i];
  mip = fmaxf(mip, 0.0f);

  float r = 0.f, g = 0.f, b = 0.f;
  { // tex0, lmax = 7
    float mm = fminf(mip, 7.0f);
    int l0 = (int)mm; float fr = mm - (float)l0; int l1 = min(l0 + 1, 7);
    bilin_acc(s_lvl0[l0], 128 >> l0, face, u, t, 1.0f - fr, r, g, b);
    bilin_acc(s_lvl0[l1], 128 >> l1, face, u, t, fr,        r, g, b);
  }
  { // tex1, lmax = 9
    float mm = fminf(mip, 9.0f);
    int l0 = (int)mm; float fr = mm - (float)l0; int l1 = min(l0 + 1, 9);
    bilin_acc(s_lvl1[l0], 512 >> l0, face, u, t, 1.0f - fr, r, g, b);
    bilin_acc(s_lvl1[l1], 512 >> l1, face, u, t, fr,        r, g, b);
  }

  float bc = fminf(fmaxf(brp[0], -1.0f), 2.0f);
  float ml = mulp[0];
  float* op = out + 3 * i;
  op[0] = fminf(fmaxf(expf(bc + ml * r), 0.01f), 1000.0f);
  op[1] = fminf(fmaxf(expf(bc + ml * g), 0.01f), 1000.0f);
  op[2] = fminf(fmaxf(expf(bc + ml * b), 0.01f), 1000.0f);
}

// ---------------------------------------------------------------------------
extern "C" void kernel_launch(void* const* d_in, const int* in_sizes, int n_in,
                              void* d_out, int out_size, void* d_ws, size_t ws_size,
                              hipStream_t stream) {
  (void)in_sizes; (void)n_in; (void)out_size; (void)ws_size;
  const float* vd  = (const float*)d_in[0];
  const float* sa  = (const float*)d_in[1];
  const float* nu  = (const float*)d_in[2];
  const float* bg0 = (const float*)d_in[3];
  const float* bg1 = (const float*)d_in[4];
  const float* brp = (const float*)d_in[5];
  const float* mlp = (const float*)d_in[6];
  const float* mbp = (const float*)d_in[7];
  float* ws  = (float*)d_ws;
  float* out = (float*)d_out;

  // Phase 1: mip pyramids (levels 1..L) into d_ws.
  {
    const float* src = bg0;
    float* dst = ws;                              // tex0: 128 -> 1 (7 levels)
    for (int bits = 6; bits >= 0; --bits) {       // Rd = 64 .. 1
      int Rd = 1 << bits;
      int N  = 6 * Rd * Rd;                       // texels
      downsample_kernel<<<(N + 255) / 256, 256, 0, stream>>>(src, dst, bits, N);
      src = dst; dst += N * 3;
    }
  }
  {
    const float* src = bg1;
    float* dst = ws + WS1_BASE;                   // tex1: 512 -> 1 (9 levels)
    for (int bits = 8; bits >= 0; --bits) {       // Rd = 256 .. 1
      int Rd = 1 << bits;
      int N  = 6 * Rd * Rd;
      downsample_kernel<<<(N + 255) / 256, 256, 0, stream>>>(src, dst, bits, N);
      src = dst; dst += N * 3;
    }
  }

  // Phase 2: sampling. 1M rays, 1024 threads/block (32 wave32 waves).
  unsigned n = B_TOTAL;
  hcubemap_sample<<<(n + 1023) / 1024, 1024, 0, stream>>>(
      vd, sa, nu, bg0, bg1, ws, brp, mlp, mbp, out, n);
}